// SOM_12850542150412
// MI455X (gfx1250) — compile-verified
//
#include <hip/hip_runtime.h>
#include <hip/hip_bf16.h>
#include <math.h>

// Problem constants (from reference)
#define D_SIZE 1024   // INPUT_SIZE (argmin axis)
#define M_SIZE 4096   // OUT_H*OUT_W (PairwiseDistance reduction axis)
#define BATCH  64
#define EPS    1e-6f

#define NQ      4            // M-quarters (cross-block split, reduced in phase 1b)
#define MQ      (M_SIZE/NQ)  // 1024 m per block
#define MW      (MQ/8)       // 128 m per wave (8 waves/block)

typedef float v2f __attribute__((ext_vector_type(2)));
typedef float v8f __attribute__((ext_vector_type(8)));

// ---------------------------------------------------------------------------
// Phase 1: partial per-row reductions of W via exact-f32 WMMA.
//   S1[d] = sum_m W[d,m],  S2[d] = sum_m W[d,m]^2
// Grid: 256 blocks (64 row-tiles x 4 M-quarters) x 8 waves -> 2048 waves so
// the 16 MB stream of W can actually saturate HBM/L2 (64 waves cannot).
// A-tile (16x4 f32 ISA layout): lane l holds row d0+(l&15), K-pair (l>>4)*2.
// The 4x16 tile equal to A^T has identical per-lane registers, so the same
// v2f feeds both operands for S2; diagonal of C = sum of squares (exact f32).
// Waves combine partials through LDS in fixed wave order (deterministic).
// ---------------------------------------------------------------------------
__global__ __launch_bounds__(256) void som_rowsum_wmma(
    const float* __restrict__ W, float* __restrict__ S1p, float* __restrict__ S2p) {
  const int lane = threadIdx.x & 31;
  const int wave = threadIdx.x >> 5;
  const int tile = blockIdx.x >> 2;            // 0..63 (16 D-rows each)
  const int q    = blockIdx.x & 3;             // M-quarter
  const int d0   = tile * 16;
  const int row  = d0 + (lane & 15);
  const int koff = (lane >> 4) << 1;           // 0 for lanes 0-15, 2 for 16-31
  const int m0   = q * MQ + wave * MW;         // this wave's 128-wide m-chunk
  const float* p = W + (size_t)row * M_SIZE + m0 + koff;

  v8f c1a = {}, c1b = {};                      // row-sum accumulators
  v8f c2a = {}, c2b = {};                      // row-sum-of-squares accumulators
  v2f ones; ones.x = 1.0f; ones.y = 1.0f;

#pragma unroll 4
  for (int m = 0; m < MW; m += 8) {
    v2f a0 = *(const v2f*)(p + m);             // K-slice m..m+3 (lane's pair)
    v2f a1 = *(const v2f*)(p + m + 4);
    c1a = __builtin_amdgcn_wmma_f32_16x16x4_f32(false, a0, false, ones, (short)0, c1a, false, false);
    c2a = __builtin_amdgcn_wmma_f32_16x16x4_f32(false, a0, false, a0,   (short)0, c2a, false, false);
    c1b = __builtin_amdgcn_wmma_f32_16x16x4_f32(false, a1, false, ones, (short)0, c1b, false, false);
    c2b = __builtin_amdgcn_wmma_f32_16x16x4_f32(false, a1, false, a1,   (short)0, c2b, false, false);
  }
  v8f c1 = c1a + c1b;
  v8f c2 = c2a + c2b;

  // C/D layout (32-bit 16x16): VGPR r, lanes 0-15 -> (M=r, N=lane);
  // lanes 16-31 -> (M=8+r, N=lane-16). Park per-wave partials in LDS.
  __shared__ float l1[8][16];
  __shared__ float l2[8][16];
#pragma unroll
  for (int i = 0; i < 8; ++i) {
    if (lane == 0)      l1[wave][i]     = c1[i];   // (M=i,   N=0)
    if (lane == 16)     l1[wave][8 + i] = c1[i];   // (M=8+i, N=0)
    if (lane == i)      l2[wave][i]     = c2[i];   // diagonal (i,i)
    if (lane == 16 + i) l2[wave][8 + i] = c2[i];   // (M=8+i, N=i), off the diagonal
  }
  __syncthreads();

  // Fixed-order (w = 0..7) cross-wave sum -> deterministic partial per quarter.
  if (threadIdx.x < 16) {
    float s1 = 0.0f, s2 = 0.0f;
#pragma unroll
    for (int w = 0; w < 8; ++w) { s1 += l1[w][threadIdx.x]; s2 += l2[w][threadIdx.x]; }
    S1p[q * D_SIZE + d0 + threadIdx.x] = s1;
    S2p[q * D_SIZE + d0 + threadIdx.x] = s2;
  }
}

// ---------------------------------------------------------------------------
// Phase 1b: fold the 4 M-quarter partials per row in fixed order.
// ---------------------------------------------------------------------------
__global__ __launch_bounds__(256) void som_reduce_partials(
    const float* __restrict__ S1p, const float* __restrict__ S2p,
    float* __restrict__ S1, float* __restrict__ S2) {
  const int d = blockIdx.x * 256 + threadIdx.x;
  float s1 = 0.0f, s2 = 0.0f;
#pragma unroll
  for (int c = 0; c < NQ; ++c) {
    s1 += S1p[c * D_SIZE + d];
    s2 += S2p[c * D_SIZE + d];
  }
  S1[d] = s1;
  S2[d] = s2;
}

// ---------------------------------------------------------------------------
// Phase 2: one block per batch element. dist^2[b,d] = M*t^2 - 2*t*S1 + S2,
// t = x[b,d]+eps. Block-wide min+argmin with first-index tie-break (matches
// jnp.argmin), gather location, stash sqrt(min) for the loss.
// ---------------------------------------------------------------------------
__global__ __launch_bounds__(256) void som_argmin(
    const float* __restrict__ x, const float* __restrict__ S1,
    const float* __restrict__ S2, const float* __restrict__ loc,
    float* __restrict__ out, float* __restrict__ minvals) {
  const int b = blockIdx.x;
  const int tid = threadIdx.x;

  float best = 3.402823466e38f;
  int   bi   = 0x7fffffff;
  for (int d = tid; d < D_SIZE; d += 256) {
    float t = x[b * D_SIZE + d] + EPS;
    float v = fmaf((float)M_SIZE * t, t, fmaf(-2.0f * t, S1[d], S2[d]));
    if (v < best) { best = v; bi = d; }        // d increases -> first-min kept
  }

  __shared__ float sv[256];
  __shared__ int   si[256];
  sv[tid] = best; si[tid] = bi;
  __syncthreads();
  for (int s = 128; s > 0; s >>= 1) {
    if (tid < s) {
      float ov = sv[tid + s]; int oi = si[tid + s];
      if (ov < sv[tid] || (ov == sv[tid] && oi < si[tid])) { sv[tid] = ov; si[tid] = oi; }
    }
    __syncthreads();
  }
  if (tid == 0) {
    int idx = si[0];
    out[2 * b]     = loc[2 * idx];
    out[2 * b + 1] = loc[2 * idx + 1];
    minvals[b] = sqrtf(fmaxf(sv[0], 0.0f));
  }
}

// ---------------------------------------------------------------------------
// Phase 3: deterministic serial sum of 64 per-batch minima -> loss.
// ---------------------------------------------------------------------------
__global__ void som_loss(const float* __restrict__ minvals, float* __restrict__ out) {
  float s = 0.0f;
  for (int i = 0; i < BATCH; ++i) s += minvals[i];
  out[2 * BATCH] = s / (float)BATCH;           // d_out[128]
}

extern "C" void kernel_launch(void* const* d_in, const int* in_sizes, int n_in,
                              void* d_out, int out_size, void* d_ws, size_t ws_size,
                              hipStream_t stream) {
  const float* x   = (const float*)d_in[0];    // [B, D]
  const float* W   = (const float*)d_in[1];    // [D, M]
  const float* loc = (const float*)d_in[2];    // [M, 2]
  float* out = (float*)d_out;                  // 128 location floats + 1 loss

  float* S1p     = (float*)d_ws;               // [NQ, D]
  float* S2p     = S1p + NQ * D_SIZE;          // [NQ, D]
  float* S1      = S2p + NQ * D_SIZE;          // [D]
  float* S2      = S1 + D_SIZE;                // [D]
  float* minvals = S2 + D_SIZE;                // [B]

  som_rowsum_wmma<<<64 * NQ, 256, 0, stream>>>(W, S1p, S2p);   // 2048 waves stream W
  som_reduce_partials<<<D_SIZE / 256, 256, 0, stream>>>(S1p, S2p, S1, S2);
  som_argmin<<<BATCH, 256, 0, stream>>>(x, S1, S2, loc, out, minvals);
  som_loss<<<1, 1, 0, stream>>>(minvals, out);
}